// LSTM_67577015435545
// MI455X (gfx1250) — compile-verified
//
#include <hip/hip_runtime.h>

// Problem constants (from the reference)
#define SEQ     1024
#define BATCH   2048
#define IN_DIM  32
#define HID     12
#define G4      48      // 4*HID gate width
// Output: (SEQ, 1) = FC over layer-1 hidden of batch element BATCH-1 at every t.

typedef __attribute__((ext_vector_type(2))) float v2f;
typedef __attribute__((ext_vector_type(8))) float v8f;

__device__ __forceinline__ float sigmoidf_(float x) {
    return 1.0f / (1.0f + __expf(-x));
}

// Single-workgroup kernel: WMMA input-projection GEMM + pipelined 2-layer recurrence.
__launch_bounds__(256, 1)
__global__ void lstm_b2047_kernel(const float* __restrict__ X,
                                  const float* __restrict__ Wih0, const float* __restrict__ Whh0,
                                  const float* __restrict__ bih0, const float* __restrict__ bhh0,
                                  const float* __restrict__ Wih1, const float* __restrict__ Whh1,
                                  const float* __restrict__ bih1, const float* __restrict__ bhh1,
                                  const float* __restrict__ Wfc,  const float* __restrict__ bfc,
                                  float* __restrict__ out,
                                  float* __restrict__ pre0 /* d_ws: SEQ*G4 floats = 192KB */)
{
    __shared__ float act0[G4];        // layer-0 gate activations for step t
    __shared__ float act1[G4];        // layer-1 gate activations for step t-1
    __shared__ float h0buf[2][HID];   // layer-0 hidden, double-buffered by t parity
    __shared__ float h1buf[HID];      // layer-1 hidden (latest)

    const int tid  = threadIdx.x;
    const int lane = tid & 31;        // wave32
    const int wave = tid >> 5;        // 8 waves
    const int l    = lane & 15;       // lane within half-wave
    const int hh   = lane >> 4;       // half-wave select

    // ---------------- Phase 1: pre0 = Xb @ Wih0^T + (bih0 + bhh0) via fp32 WMMA -------
    // Xb[t][c] = X[t, BATCH-1, c].  GEMM: M=1024 (64 tiles), N=48 (3 tiles), K=32 (8x4).
    const long xoff = (long)(BATCH - 1) * IN_DIM;
    for (int tile = wave; tile < 64 * 3; tile += 8) {       // uniform per wave -> EXEC all 1s
        const int m   = tile / 3;
        const int n   = tile % 3;
        const int col = n * 16 + l;                          // N index owned by this lane
        const float bsum = bih0[col] + bhh0[col];            // bias folded into C init
        v8f acc;
        #pragma unroll
        for (int r = 0; r < 8; ++r) acc[r] = bsum;           // C[r + 8h][col]

        const int   row  = m * 16 + l;                       // M index (A layout)
        const float* xrow = X + (long)row * (BATCH * IN_DIM) + xoff;
        #pragma unroll
        for (int k = 0; k < 8; ++k) {
            const int k0 = k * 4 + 2 * hh;                   // A 16x4: a[r] = A[row][K = 2h + r]
            v2f a, b;
            a.x = xrow[k0];
            a.y = xrow[k0 + 1];
            // B 4x16: b[r] = B[K = 2h + r][col],  B = Wih0^T  ->  Wih0[col][K]
            const float* wrow = Wih0 + col * IN_DIM + k0;
            b.x = wrow[0];
            b.y = wrow[1];
            acc = __builtin_amdgcn_wmma_f32_16x16x4_f32(
                      /*neg_a=*/false, a, /*neg_b=*/false, b,
                      /*c_mod=*/(short)0, acc, /*reuse_a=*/false, /*reuse_b=*/false);
        }
        #pragma unroll
        for (int r = 0; r < 8; ++r)                          // C 16x16: M = r + 8h, N = col
            pre0[(m * 16 + r + 8 * hh) * G4 + col] = acc[r];
    }

    // ---------------- Per-lane weight registers for the recurrence --------------------
    float wr0[HID];                   // lanes 0..47:   Whh0 row
    float wi1[HID], wr1[HID];         // lanes 64..111: Wih1 / Whh1 rows
    float wfc[HID];                   // lane 127:      Wfc row
    float b1 = 0.0f, c0 = 0.0f, c1 = 0.0f, bo = 0.0f;
    const int j1 = tid - 64;

    if (tid < G4) {
        #pragma unroll
        for (int k = 0; k < HID; ++k) wr0[k] = Whh0[tid * HID + k];
    }
    if (tid >= 64 && tid < 64 + G4) {
        #pragma unroll
        for (int k = 0; k < HID; ++k) { wi1[k] = Wih1[j1 * HID + k]; wr1[k] = Whh1[j1 * HID + k]; }
        b1 = bih1[j1] + bhh1[j1];
    }
    if (tid == 127) {
        #pragma unroll
        for (int k = 0; k < HID; ++k) wfc[k] = Wfc[k];
        bo = bfc[0];
    }
    if (tid < HID) { h0buf[0][tid] = 0.0f; h0buf[1][tid] = 0.0f; h1buf[tid] = 0.0f; }
    __syncthreads();                  // pre0 complete + state initialized

    // ---------------- Phase 2: pipelined recurrence -----------------------------------
    // Iteration t: layer0 computes step t (t<SEQ); layer1 computes step t-1 (t>=1);
    // lane 127 emits out[t-2] (t>=2).  pre0[t] is register-prefetched one step ahead.
    float pcur = (tid < G4) ? pre0[tid] : 0.0f;

    for (int t = 0; t <= SEQ; ++t) {
        // ---- stage 1: gate evaluation (reads h state, writes activations) ----
        if (tid < G4 && t < SEQ) {
            float z = pcur;
            const float* hp = h0buf[(t + 1) & 1];            // h0[t-1]
            #pragma unroll
            for (int k = 0; k < HID; ++k) z += wr0[k] * hp[k];
            act0[tid] = (tid >= 2 * HID && tid < 3 * HID) ? tanhf(z) : sigmoidf_(z);
        }
        if (tid >= 64 && tid < 64 + G4 && t >= 1) {
            float z = b1;
            const float* hp = h0buf[(t + 1) & 1];            // h0[t-1] = layer-1 input at step t-1
            #pragma unroll
            for (int k = 0; k < HID; ++k) z += wi1[k] * hp[k] + wr1[k] * h1buf[k]; // h1[t-2]
            act1[j1] = (j1 >= 2 * HID && j1 < 3 * HID) ? tanhf(z) : sigmoidf_(z);
        }
        if (tid == 127 && t >= 2) {                          // h1buf currently holds h1[t-2]
            float o = bo;
            #pragma unroll
            for (int k = 0; k < HID; ++k) o += wfc[k] * h1buf[k];
            out[t - 2] = o;
        }
        float pnext = (tid < G4 && t + 1 < SEQ) ? pre0[(t + 1) * G4 + tid] : 0.0f;
        __syncthreads();

        // ---- stage 2: cell/hidden update by owner lanes ----
        if (tid < HID && t < SEQ) {
            const float si = act0[tid], sf = act0[HID + tid];
            const float tg = act0[2 * HID + tid], so = act0[3 * HID + tid];
            c0 = sf * c0 + si * tg;
            h0buf[t & 1][tid] = so * tanhf(c0);              // h0[t]
        }
        if (tid >= 64 && tid < 64 + HID && t >= 1) {
            const int j = tid - 64;
            const float si = act1[j], sf = act1[HID + j];
            const float tg = act1[2 * HID + j], so = act1[3 * HID + j];
            c1 = sf * c1 + si * tg;
            h1buf[j] = so * tanhf(c1);                       // h1[t-1]
        }
        pcur = pnext;
        __syncthreads();
    }

    if (tid == 127) {                                        // final output: h1[SEQ-1]
        float o = bo;
        #pragma unroll
        for (int k = 0; k < HID; ++k) o += wfc[k] * h1buf[k];
        out[SEQ - 1] = o;
    }
}

extern "C" void kernel_launch(void* const* d_in, const int* in_sizes, int n_in,
                              void* d_out, int out_size, void* d_ws, size_t ws_size,
                              hipStream_t stream) {
    (void)in_sizes; (void)n_in; (void)out_size; (void)ws_size;
    const float* X    = (const float*)d_in[0];
    const float* Wih0 = (const float*)d_in[1];
    const float* Whh0 = (const float*)d_in[2];
    const float* bih0 = (const float*)d_in[3];
    const float* bhh0 = (const float*)d_in[4];
    const float* Wih1 = (const float*)d_in[5];
    const float* Whh1 = (const float*)d_in[6];
    const float* bih1 = (const float*)d_in[7];
    const float* bhh1 = (const float*)d_in[8];
    const float* Wfc  = (const float*)d_in[9];
    const float* bfc  = (const float*)d_in[10];
    float* out  = (float*)d_out;
    float* pre0 = (float*)d_ws;   // needs SEQ*G4*4 = 192KB of scratch

    // Latency-bound sequential recurrence: one workgroup (8 wave32s) owns everything.
    lstm_b2047_kernel<<<1, 256, 0, stream>>>(X, Wih0, Whh0, bih0, bhh0,
                                             Wih1, Whh1, bih1, bhh1, Wfc, bfc,
                                             out, pre0);
}